// DSQGAttentionD41J16D_13305808683381
// MI455X (gfx1250) — compile-verified
//
#include <hip/hip_runtime.h>
#include <math.h>

typedef float v2f __attribute__((ext_vector_type(2)));
typedef float v4f __attribute__((ext_vector_type(4)));
typedef float v8f __attribute__((ext_vector_type(8)));

#define HH 16
#define NN 4096
#define HD 64
#define NEG_BIG (-1e30f)

// One wave32 handles one (head, 16-query tile).
// Lane pair (m, m+16) co-owns query row m = lane&15; hi = lane>>4 selects
// the half of the 64-dim head (and the half of the 16 offsets in softmax).
__global__ __launch_bounds__(128) void dsqg_attn_kernel(
    const float* __restrict__ gq, const float* __restrict__ gk,
    const float* __restrict__ gv, const float* __restrict__ gpb,
    const float* __restrict__ gse, float* __restrict__ gout)
{
    constexpr int offs[16] = {1, 3, 4, 13, 15, 21, 23, 28,
                              48, 64, 96, 192, 384, 512, 768, 1024};
    const float sc = 0.125f;  // 1/sqrt(64)

    const int lane = threadIdx.x & 31;
    const int wave = threadIdx.x >> 5;
    const int row  = lane & 15;   // query-in-tile m (and SE row / B column)
    const int hi   = lane >> 4;   // 0 or 1

    const int bx = blockIdx.x;            // 1024 blocks: 16 heads x 64 blocks
    const int h  = bx >> 6;
    const int n0 = ((bx & 63) * 4 + wave) * 16;
    const int nq = n0 + row;              // this lane-pair's query index

    const size_t head_base = (size_t)h * NN * HD;
    const float* qrow = gq + head_base + (size_t)nq * HD;

    // ---------------- Q fragments in WMMA-B layout (4x16 chunks) -----------
    // chunk c covers dims [4c, 4c+4): lanes<16 hold K=0,1; lanes>=16 hold K=2,3
    v2f qf[16];
#pragma unroll
    for (int c = 0; c < 16; ++c)
        qf[c] = *(const v2f*)(qrow + 4 * c + 2 * hi);

    // ---------------- scale_embed GEMM via V_WMMA_F32_16X16X4_F32 ----------
    // D[i][m] = sum_d se[i][d] * q[m][d]; A = scale_embed (16x64), B = Q^T.
    // A chunk: lane holds row i=row, same K split as B.
    v8f accS = {};
    {
        const float* serow = gse + row * HD;
#pragma unroll
        for (int c = 0; c < 16; ++c) {
            v2f af = *(const v2f*)(serow + 4 * c + 2 * hi);
            accS = __builtin_amdgcn_wmma_f32_16x16x4_f32(
                false, af, false, qf[c], (short)0, accS, false, false);
        }
    }
    // C-layout result: lane (m=row, hi) holds accS[r] = se[i].q[m], i = r+8*hi

    // ---------------- linear Q copy for VALU band-dots ---------------------
    // lane (row,hi) owns dims [32*hi, 32*hi+32): contiguous b128 loads.
    v4f qlin[8];
#pragma unroll
    for (int c = 0; c < 8; ++c)
        qlin[c] = *(const v4f*)(qrow + 32 * hi + 4 * c);

    // ---------------- QK band dots + per-offset scores ---------------------
    float s_own[8];  // scores for this lane's 8 owned offsets (i = r + 8*hi)
#pragma unroll
    for (int i = 0; i < 16; ++i) {
        const int nk     = nq - offs[i];
        const bool valid = (nk >= 0);
        const int nkc    = valid ? nk : 0;
        const float* krow = gk + head_base + (size_t)nkc * HD + 32 * hi;
        // independent per-component accumulation -> v_pk_fma_f32 packing
        v4f acc = {0.f, 0.f, 0.f, 0.f};
#pragma unroll
        for (int c = 0; c < 8; ++c) {
            v4f kf = *(const v4f*)(krow + 4 * c);
            acc += qlin[c] * kf;
        }
        float part = (acc.x + acc.y) + (acc.z + acc.w);
        // combine the two 32-dim halves held by lanes m and m+16
        float dot = part + __shfl_xor(part, 16, 32);
        if ((i >> 3) == hi) {
            const int r = i & 7;
            s_own[r] = valid
                ? fmaf(dot + accS[r], sc, gpb[i * HH + h])
                : NEG_BIG;
        }
    }

    // ---------------- softmax over 16 offsets (8 per lane of the pair) -----
    float m8 = s_own[0];
#pragma unroll
    for (int r = 1; r < 8; ++r) m8 = fmaxf(m8, s_own[r]);
    const float mAll = fmaxf(m8, __shfl_xor(m8, 16, 32));

    float p_own[8];
    float l8 = 0.f;
#pragma unroll
    for (int r = 0; r < 8; ++r) {
        const float sv = s_own[r];
        const float pr = (sv > -1e29f) ? __expf(sv - mAll) : 0.f;
        p_own[r] = pr;
        l8 += pr;
    }
    const float l  = l8 + __shfl_xor(l8, 16, 32);
    const float rl = (l > 0.f) ? (1.f / l) : 1.f;  // matches where(l>0,l,1)

    // exchange p so each lane has all 16 probabilities
    float p_all[16];
#pragma unroll
    for (int r = 0; r < 8; ++r) {
        const float po = p_own[r];
        const float px = __shfl_xor(po, 16, 32);
        p_all[r]     = hi ? px : po;
        p_all[r + 8] = hi ? po : px;
    }

    // ---------------- PV gather-accumulate (packed FMA) --------------------
    v4f o[8];
#pragma unroll
    for (int c = 0; c < 8; ++c) o[c] = (v4f){0.f, 0.f, 0.f, 0.f};

#pragma unroll
    for (int i = 0; i < 16; ++i) {
        const int nk   = nq - offs[i];
        const int nkc  = (nk >= 0) ? nk : 0;
        const float pi = p_all[i];  // 0 for invalid -> clamped row contributes nothing
        const float* vrow = gv + head_base + (size_t)nkc * HD + 32 * hi;
#pragma unroll
        for (int c = 0; c < 8; ++c) {
            v4f vf = *(const v4f*)(vrow + 4 * c);
            o[c] += vf * pi;   // vector fma with scalar broadcast
        }
    }

    float* orow = gout + head_base + (size_t)nq * HD + 32 * hi;
#pragma unroll
    for (int c = 0; c < 8; ++c) {
        v4f val = o[c] * rl;
        *(v4f*)(orow + 4 * c) = val;
    }
}

extern "C" void kernel_launch(void* const* d_in, const int* in_sizes, int n_in,
                              void* d_out, int out_size, void* d_ws, size_t ws_size,
                              hipStream_t stream) {
    const float* q  = (const float*)d_in[0];
    const float* k  = (const float*)d_in[1];
    const float* v  = (const float*)d_in[2];
    const float* pb = (const float*)d_in[3];
    const float* se = (const float*)d_in[4];
    float* out = (float*)d_out;

    // 16 heads x 64 blocks; each block = 4 waves x 16 queries = 64 queries.
    dim3 grid(1024), block(128);
    hipLaunchKernelGGL(dsqg_attn_kernel, grid, block, 0, stream,
                       q, k, v, pb, se, out);
}